// TrulyNeuralUARTV2_32512902431191
// MI455X (gfx1250) — compile-verified
//
#include <hip/hip_runtime.h>
#include <math.h>

// Problem constants (match reference)
#define NBUF 128   // buffer_size
#define NBITS 32   // byte_bits
#define NKD  64    // key_dim

typedef __attribute__((ext_vector_type(2))) float v2f;
typedef __attribute__((ext_vector_type(8))) float v8f;

__device__ __forceinline__ float gelu_exact(float x) {
    // jax.nn.gelu(approximate=False): 0.5*x*(1+erf(x/sqrt(2)))
    return 0.5f * x * (1.0f + erff(x * 0.70710678118654752440f));
}

// One 16x16 f32 tile of D = A * B accumulated over K in steps of 4 using
// V_WMMA_F32_16X16X4_F32.  Layouts per CDNA5 ISA 7.12.2:
//   A (16x4 f32): lanes 0-15 -> M=lane, VGPR{0,1}=K{0,1}; lanes 16-31 -> K{2,3}
//   B (4x16 f32): row-striped across lanes; VGPR j, half h -> K = 2*h + j, N = lane&15
//   C/D (16x16 f32): VGPR r -> M = r + 8*half, N = lane&15
__device__ __forceinline__ v8f wmma_tile_f32(
    const float* __restrict__ A, int lda, bool transA,
    const float* __restrict__ Bm, int ldb, bool transB,
    int mBase, int nBase, int K, int lane)
{
    v8f acc = {0.f, 0.f, 0.f, 0.f, 0.f, 0.f, 0.f, 0.f};
    const int half = lane >> 4;
    const int mn   = lane & 15;
    for (int k0 = 0; k0 < K; k0 += 4) {
        const int ka = k0 + 2 * half;
        v2f a, b;
        if (transA) {   // logical A[m][k] = A[k*lda + m]
            a[0] = A[(ka    ) * lda + mBase + mn];
            a[1] = A[(ka + 1) * lda + mBase + mn];
        } else {        // logical A[m][k] = A[m*lda + k]
            a[0] = A[(mBase + mn) * lda + ka];
            a[1] = A[(mBase + mn) * lda + ka + 1];
        }
        if (transB) {   // logical B[k][n] = Bm[n*ldb + k]
            b[0] = Bm[(nBase + mn) * ldb + ka];
            b[1] = Bm[(nBase + mn) * ldb + ka + 1];
        } else {        // logical B[k][n] = Bm[k*ldb + n]
            b[0] = Bm[(ka    ) * ldb + nBase + mn];
            b[1] = Bm[(ka + 1) * ldb + nBase + mn];
        }
        acc = __builtin_amdgcn_wmma_f32_16x16x4_f32(
            false, a, false, b, (short)0, acc, false, false);
    }
    return acc;
}

// ---------------------------------------------------------------------------
// Kernel 0: zero the accumulator region of workspace.
__global__ void zero_ws_kernel(float* __restrict__ ws, int n) {
    int i = blockIdx.x * blockDim.x + threadIdx.x;
    if (i < n) ws[i] = 0.0f;
}

// ---------------------------------------------------------------------------
// Kernel 1: per-position histogram.  One wave streams one byte_bits row
// (32 floats = 128B coalesced) at a time; LDS float atomics, then one
// global float-atomic flush per block.
__global__ void hist_kernel(const int* __restrict__ pos,
                            const float* __restrict__ bits,
                            float* __restrict__ gS,    // [NBUF*NBITS]
                            float* __restrict__ gCnt,  // [NBUF]
                            int nrows)
{
    __shared__ float sS[NBUF * NBITS];
    __shared__ float sC[NBUF];
    for (int i = threadIdx.x; i < NBUF * NBITS; i += blockDim.x) sS[i] = 0.0f;
    for (int i = threadIdx.x; i < NBUF;         i += blockDim.x) sC[i] = 0.0f;
    __syncthreads();

    const int lane   = threadIdx.x & 31;
    const int wave   = threadIdx.x >> 5;
    const int wPerB  = blockDim.x >> 5;
    const int row0   = blockIdx.x * wPerB + wave;
    const int stride = gridDim.x * wPerB;

    for (int b = row0; b < nrows; b += stride) {
        int p = pos[b];                         // uniform within wave
        float v = bits[(long long)b * NBITS + lane];
        if (b + stride < nrows)
            __builtin_prefetch(&bits[(long long)(b + stride) * NBITS + lane], 0, 0);
        atomicAdd(&sS[p * NBITS + lane], v);
        if (lane == 0) atomicAdd(&sC[p], 1.0f);
    }
    __syncthreads();
    for (int i = threadIdx.x; i < NBUF * NBITS; i += blockDim.x) atomicAdd(&gS[i], sS[i]);
    for (int i = threadIdx.x; i < NBUF;         i += blockDim.x) atomicAdd(&gCnt[i], sC[i]);
}

// ---------------------------------------------------------------------------
// Kernel 2: the whole collapsed dense block, one workgroup (256 thr = 8 waves).
// Dynamic LDS layout (floats):
//   Hs   [0     .. 8192 )  : H[128][64]      (reused as SIG[128][32] in stage 6/7)
//   Qs   [8192  .. 16384)  : Q[128][64]      (reused: S[128][32] @8192, WUM[128][32] @12288)
//   As   [16384 .. 32768)  : A[128][128] logits -> softmax(attention)
//   cnt  [32768 .. 32896)
//   mask [32896 .. 33024)
__global__ void compute_kernel(const float* __restrict__ W1,
                               const float* __restrict__ b1,
                               const float* __restrict__ W2,
                               const float* __restrict__ b2,
                               const float* __restrict__ tx_keys,
                               const float* __restrict__ tx_buffer,
                               const float* __restrict__ temperature,
                               const float* __restrict__ write_lr,
                               const float* __restrict__ gS,
                               const float* __restrict__ gCnt,
                               float* __restrict__ gR,
                               int nrows)
{
    extern __shared__ float smem[];
    float* Hs    = smem;             // 128*64
    float* Qs    = smem + 8192;      // 128*64
    float* As    = smem + 16384;     // 128*128
    float* Ss    = Qs;               // 128*32   (after stage 3)
    float* WUMs  = Qs + 4096;        // 128*32
    float* SIGs  = Hs;               // 128*32   (after stage 2)
    float* cnt_s = smem + 32768;     // 128
    float* mask_s= smem + 32896;     // 128

    const int tid  = threadIdx.x;
    const int lane = tid & 31;
    const int wave = tid >> 5;
    const float invB = 1.0f / (float)nrows;

    // ---- Stage 1: H = gelu(W1 + b1)  [128x64] ----
    for (int i = tid; i < NBUF * NKD; i += blockDim.x) {
        int j = i & (NKD - 1);
        Hs[i] = gelu_exact(W1[i] + b1[j]);
    }
    __syncthreads();

    // ---- Stage 2: Q = H @ W2 + b2  [128x64] : 8x4 = 32 tiles ----
    for (int t = wave; t < 32; t += 8) {
        int mt = t >> 2, nt = t & 3;
        v8f acc = wmma_tile_f32(Hs, NKD, false, W2, NKD, false,
                                mt * 16, nt * 16, NKD, lane);
        int half = lane >> 4, mn = lane & 15;
        int col = nt * 16 + mn;
        float bias = b2[col];
        #pragma unroll
        for (int r = 0; r < 8; ++r)
            Qs[(mt * 16 + r + 8 * half) * NKD + col] = acc[r] + bias;
    }
    __syncthreads();

    // ---- Stage 3: key_sim = Q @ tx_keys^T  [128x128] : 8x8 = 64 tiles ----
    for (int t = wave; t < 64; t += 8) {
        int mt = t >> 3, nt = t & 7;
        v8f acc = wmma_tile_f32(Qs, NKD, false, tx_keys, NKD, true,
                                mt * 16, nt * 16, NKD, lane);
        int half = lane >> 4, mn = lane & 15;
        #pragma unroll
        for (int r = 0; r < 8; ++r)
            As[(mt * 16 + r + 8 * half) * NBUF + nt * 16 + mn] = acc[r];
    }
    __syncthreads();

    // ---- Softmax over each row of As (with temperature) ----
    {
        float temp = fmaxf(fabsf(temperature[0]), 0.1f);
        float invt = 1.0f / temp;
        for (int row = wave; row < NBUF; row += 8) {
            float* rp = As + row * NBUF;
            float x0 = rp[lane     ] * invt;
            float x1 = rp[lane + 32] * invt;
            float x2 = rp[lane + 64] * invt;
            float x3 = rp[lane + 96] * invt;
            float mx = fmaxf(fmaxf(x0, x1), fmaxf(x2, x3));
            #pragma unroll
            for (int off = 16; off > 0; off >>= 1)
                mx = fmaxf(mx, __shfl_xor(mx, off, 32));
            float e0 = expf(x0 - mx), e1 = expf(x1 - mx);
            float e2 = expf(x2 - mx), e3 = expf(x3 - mx);
            float s = e0 + e1 + e2 + e3;
            #pragma unroll
            for (int off = 16; off > 0; off >>= 1)
                s += __shfl_xor(s, off, 32);
            float inv = 1.0f / s;
            rp[lane     ] = e0 * inv;
            rp[lane + 32] = e1 * inv;
            rp[lane + 64] = e2 * inv;
            rp[lane + 96] = e3 * inv;
        }
        // Stage 4 (overlapped, touches Qs/cnt only): load S and counts
        for (int i = tid; i < NBUF * NBITS; i += blockDim.x) Ss[i] = gS[i];
        for (int i = tid; i < NBUF;         i += blockDim.x) cnt_s[i] = gCnt[i];
    }
    __syncthreads();

    // ---- mask[p] = (1/B) * sum_v cnt[v] * A[v][p] ----
    if (tid < NBUF) {
        float acc = 0.0f;
        for (int v = 0; v < NBUF; ++v) acc += cnt_s[v] * As[v * NBUF + tid];
        mask_s[tid] = acc * invB;
    }
    // ---- Stage 5: WUM = (A^T @ S) / B  [128x32] : 8x2 = 16 tiles ----
    for (int t = wave; t < 16; t += 8) {
        int mt = t >> 1, nt = t & 1;
        v8f acc = wmma_tile_f32(As, NBUF, true, Ss, NBITS, false,
                                mt * 16, nt * 16, NBUF, lane);
        int half = lane >> 4, mn = lane & 15;
        #pragma unroll
        for (int r = 0; r < 8; ++r)
            WUMs[(mt * 16 + r + 8 * half) * NBITS + nt * 16 + mn] = acc[r] * invB;
    }
    __syncthreads();

    // ---- Stage 6: new_buffer -> sigmoid  [128x32] ----
    {
        float lr = fminf(fmaxf(fabsf(write_lr[0]), 0.01f), 1.0f);
        for (int i = tid; i < NBUF * NBITS; i += blockDim.x) {
            int p = i >> 5;
            float c  = fminf(fmaxf(WUMs[i], 0.01f), 0.99f);
            float bl = logf(c / (1.0f - c));
            float mk = mask_s[p];
            float nb = (1.0f - lr * mk) * tx_buffer[i] + lr * mk * bl;
            SIGs[i] = 1.0f / (1.0f + expf(-nb));
        }
    }
    __syncthreads();

    // ---- Stage 7: R = A @ sigmoid(new_buffer)  [128x32] : 16 tiles ----
    for (int t = wave; t < 16; t += 8) {
        int mt = t >> 1, nt = t & 1;
        v8f acc = wmma_tile_f32(As, NBUF, false, SIGs, NBITS, false,
                                mt * 16, nt * 16, NBUF, lane);
        int half = lane >> 4, mn = lane & 15;
        #pragma unroll
        for (int r = 0; r < 8; ++r)
            gR[(mt * 16 + r + 8 * half) * NBITS + nt * 16 + mn] = acc[r];
    }
}

// ---------------------------------------------------------------------------
// Kernel 3: out[b, :] = R[pos[b], :]   (pure bandwidth: 32 MB write)
__global__ void gather_kernel(const int* __restrict__ pos,
                              const float* __restrict__ R,
                              float* __restrict__ out,
                              int nrows)
{
    long long i = (long long)blockIdx.x * blockDim.x + threadIdx.x;
    int b = (int)(i >> 5);
    int d = (int)(i & 31);
    if (b < nrows) {
        int p = pos[b];                  // uniform across the 32 lanes of a row
        out[i] = R[p * NBITS + d];
    }
}

// ---------------------------------------------------------------------------
extern "C" void kernel_launch(void* const* d_in, const int* in_sizes, int n_in,
                              void* d_out, int out_size, void* d_ws, size_t ws_size,
                              hipStream_t stream)
{
    const int*   pos  = (const int*)  d_in[0];
    const float* bits = (const float*)d_in[1];
    const float* W1   = (const float*)d_in[2];
    const float* b1   = (const float*)d_in[3];
    const float* W2   = (const float*)d_in[4];
    const float* b2   = (const float*)d_in[5];
    const float* txk  = (const float*)d_in[6];
    const float* txb  = (const float*)d_in[7];
    const float* temp = (const float*)d_in[8];
    const float* wlr  = (const float*)d_in[9];

    float* ws  = (float*)d_ws;
    float* gS  = ws;                 // 4096 floats
    float* gC  = ws + 4096;          // 128 floats
    float* gR  = ws + 4224;          // 4096 floats

    const int B = in_sizes[0];       // 262144 rows

    // 1) zero accumulators (S + count)
    zero_ws_kernel<<<(4224 + 255) / 256, 256, 0, stream>>>(ws, 4224);

    // 2) per-position histogram over byte_bits
    hist_kernel<<<512, 256, 0, stream>>>(pos, bits, gS, gC, B);

    // 3) dense 128x* block: attention matrix, buffer update, R = A @ sig(buf)
    size_t lds_bytes = (size_t)33024 * sizeof(float);   // ~129 KB < 320 KB/WGP
    compute_kernel<<<1, 256, lds_bytes, stream>>>(W1, b1, W2, b2, txk, txb,
                                                  temp, wlr, gS, gC, gR, B);

    // 4) gather output rows
    long long total = (long long)B * NBITS;
    int blocks = (int)((total + 255) / 256);
    gather_kernel<<<blocks, 256, 0, stream>>>(pos, gR, (float*)d_out, B);
}